// ConvSkip_41360535061062
// MI455X (gfx1250) — compile-verified
//
#include <hip/hip_runtime.h>

// ---------------------------------------------------------------------------
// ConvSkip graph-conv layer for MI455X (gfx1250, wave32, WMMA).
//
// out = relu( lap + merge @ W_tr^T + b_tr )
//   lin = data @ W_lin^T + b_lin
//   lap[i] = deg[i] > 0 ? lin[i] - (sum_{e: tgt=i} lin[src[e]]) / deg[i] : 0
// (the lin[tgt] terms of the reference telescope to exactly lin[i], so only
//  one gather+scatter per edge is needed)
//
// Cost model: GEMMs are 0.8 GFLOP (noise; done f32-exact with
// V_WMMA_F32_16X16X4_F32). Edge traffic is ~410 MB of gathers + f32 atomic
// adds, but lin/acc (12.8 MB each) are L2-resident (192 MB L2), so the
// kernel is bounded by L2 atomic throughput, not the 23.3 TB/s HBM.
// Atomics are forced to native global_atomic_add_f32 via inline asm.
// ---------------------------------------------------------------------------

typedef float v2f __attribute__((ext_vector_type(2)));
typedef float v4f __attribute__((ext_vector_type(4)));
typedef float v8f __attribute__((ext_vector_type(8)));

#define NN 50000
#define NE 800000
#define DD 64
#define NTILES (NN / 16)   // 3125 exactly -> no ragged tiles, EXEC all-1s

// Non-returning native f32 atomic add (STOREcnt-tracked; kernel-end waits).
__device__ __forceinline__ void atomic_add_f32(float* p, float v) {
  asm volatile("global_atomic_add_f32 %0, %1, off"
               :
               : "v"(p), "v"(v)
               : "memory");
}

// ---- workspace zeroing ----------------------------------------------------
__global__ void zero_ws_kernel(float* __restrict__ p, int n) {
  int i = blockIdx.x * blockDim.x + threadIdx.x;
  if (i < n) p[i] = 0.0f;
}

// ---- degree histogram: deg[tgt[e]] += 1 -----------------------------------
__global__ void deg_kernel(const int* __restrict__ tgt, float* __restrict__ deg) {
  int e = blockIdx.x * blockDim.x + threadIdx.x;
  if (e < NE) atomic_add_f32(&deg[tgt[e]], 1.0f);
}

// ---- GEMM 1: lin = data @ W_lin^T + b_lin  (WMMA f32 16x16x4) -------------
// A(16x4) lane layout: lane = m + 16*(k/2), VGPR v holds K = 2*(lane>>4)+v
// B(4x16) lane layout: lane = n + 16*(k/2);  B[k][n] = W[n][k] -> contiguous
// C(16x16): VGPR r holds M = r + 8*(lane>>4), N = lane&15
__global__ void lin_gemm_kernel(const float* __restrict__ data,
                                const float* __restrict__ W,
                                const float* __restrict__ bias,
                                float* __restrict__ lin) {
  const int wave = threadIdx.x >> 5;
  const int tile = blockIdx.x * (blockDim.x >> 5) + wave;
  if (tile >= NTILES) return;                 // uniform per-wave branch
  const int lane = threadIdx.x & 31;
  const int hi = lane >> 4;                   // 0 | 1 (K half selector)
  const int lo = lane & 15;                   // M (for A) / N (for B,C)

  // A: 16 K-steps of float2, row m = lo of this tile
  const float* arow = data + (size_t)(tile * 16 + lo) * DD + 2 * hi;
  v2f a[16];
#pragma unroll
  for (int ks = 0; ks < 16; ++ks) a[ks] = *(const v2f*)(arow + ks * 4);

#pragma unroll
  for (int ct = 0; ct < 4; ++ct) {            // 4 column tiles of 16
    const int n = ct * 16 + lo;
    const float* brow = W + (size_t)n * DD + 2 * hi;   // W row n, K-contig
    v8f c = {};
#pragma unroll
    for (int ks = 0; ks < 16; ++ks) {
      v2f bv = *(const v2f*)(brow + ks * 4);
      c = __builtin_amdgcn_wmma_f32_16x16x4_f32(
          /*neg_a=*/false, a[ks], /*neg_b=*/false, bv,
          /*c_mod=*/(short)0, c, /*reuse_a=*/false, /*reuse_b=*/false);
    }
    const float bn = bias[n];
#pragma unroll
    for (int r = 0; r < 8; ++r) {
      const int m = tile * 16 + r + 8 * hi;
      lin[(size_t)m * DD + n] = c[r] + bn;    // lanes 0..15 contiguous in n
    }
  }
}

// ---- edge scatter: acc[tgt[e]][:] += lin[src[e]][:] -----------------------
// 16 threads per edge, 4 features each: one b128 gather + 4 native f32
// atomics per thread. 51.2M atomics total, all L2-resident.
__global__ void scatter_kernel(const int* __restrict__ src,
                               const int* __restrict__ tgt,
                               const float* __restrict__ lin,
                               float* __restrict__ acc) {
  const long long tid = (long long)blockIdx.x * blockDim.x + threadIdx.x;
  const int e = (int)(tid >> 4);
  const int d = (int)(tid & 15) * 4;
  if (e < NE) {
    const v4f v = *(const v4f*)(lin + (size_t)src[e] * DD + d);
    float* dst = acc + (size_t)tgt[e] * DD + d;
    atomic_add_f32(dst + 0, v.x);
    atomic_add_f32(dst + 1, v.y);
    atomic_add_f32(dst + 2, v.z);
    atomic_add_f32(dst + 3, v.w);
  }
}

// ---- GEMM 2 + fused epilogue: out = relu(merge@W_tr^T + b_tr + lap) -------
__global__ void out_gemm_kernel(const float* __restrict__ merge,
                                const float* __restrict__ W,
                                const float* __restrict__ bias,
                                const float* __restrict__ lin,
                                const float* __restrict__ acc,
                                const float* __restrict__ deg,
                                float* __restrict__ out) {
  const int wave = threadIdx.x >> 5;
  const int tile = blockIdx.x * (blockDim.x >> 5) + wave;
  if (tile >= NTILES) return;
  const int lane = threadIdx.x & 31;
  const int hi = lane >> 4;
  const int lo = lane & 15;

  const float* arow = merge + (size_t)(tile * 16 + lo) * DD + 2 * hi;
  v2f a[16];
#pragma unroll
  for (int ks = 0; ks < 16; ++ks) a[ks] = *(const v2f*)(arow + ks * 4);

  // deg for the 8 rows this lane's C fragment touches (independent of ct)
  float dg[8];
#pragma unroll
  for (int r = 0; r < 8; ++r) dg[r] = deg[tile * 16 + r + 8 * hi];

#pragma unroll
  for (int ct = 0; ct < 4; ++ct) {
    const int n = ct * 16 + lo;
    const float* brow = W + (size_t)n * DD + 2 * hi;
    v8f c = {};
#pragma unroll
    for (int ks = 0; ks < 16; ++ks) {
      v2f bv = *(const v2f*)(brow + ks * 4);
      c = __builtin_amdgcn_wmma_f32_16x16x4_f32(
          false, a[ks], false, bv, (short)0, c, false, false);
    }
    const float bn = bias[n];
#pragma unroll
    for (int r = 0; r < 8; ++r) {
      const int m = tile * 16 + r + 8 * hi;
      const size_t off = (size_t)m * DD + n;
      float lap = 0.0f;
      if (dg[r] > 0.0f) lap = lin[off] - acc[off] / dg[r];
      const float v = c[r] + bn + lap;
      out[off] = v > 0.0f ? v : 0.0f;
    }
  }
}

// ---------------------------------------------------------------------------
extern "C" void kernel_launch(void* const* d_in, const int* in_sizes, int n_in,
                              void* d_out, int out_size, void* d_ws, size_t ws_size,
                              hipStream_t stream) {
  const float* data  = (const float*)d_in[0];
  const float* merge = (const float*)d_in[1];
  const int*   src   = (const int*)d_in[2];
  const int*   tgt   = (const int*)d_in[3];
  const float* W_lin = (const float*)d_in[4];
  const float* b_lin = (const float*)d_in[5];
  const float* W_tr  = (const float*)d_in[6];
  const float* b_tr  = (const float*)d_in[7];
  float* out = (float*)d_out;

  // workspace partition: lin[N*64] | acc[N*64] | deg[N]   (~25.8 MB)
  float* lin = (float*)d_ws;
  float* acc = lin + (size_t)NN * DD;
  float* deg = acc + (size_t)NN * DD;

  // 1) zero acc+deg (contiguous region) every call — graph-capture safe
  {
    const int n = NN * DD + NN;
    zero_ws_kernel<<<(n + 255) / 256, 256, 0, stream>>>(acc, n);
  }
  // 2) degree histogram
  deg_kernel<<<(NE + 255) / 256, 256, 0, stream>>>(tgt, deg);
  // 3) lin = data @ W_lin^T + b_lin   (4 waves/block, 1 tile/wave)
  {
    const int waves_per_block = 4;
    const int blocks = (NTILES + waves_per_block - 1) / waves_per_block;
    lin_gemm_kernel<<<blocks, waves_per_block * 32, 0, stream>>>(data, W_lin, b_lin, lin);
  }
  // 4) scatter-add lin[src] into acc[tgt] : E*16 threads, 4 feats each
  {
    const long long total = (long long)NE * 16;
    scatter_kernel<<<(int)(total / 256), 256, 0, stream>>>(src, tgt, lin, acc);
  }
  // 5) fused GEMM2 + laplacian + relu
  {
    const int waves_per_block = 4;
    const int blocks = (NTILES + waves_per_block - 1) / waves_per_block;
    out_gemm_kernel<<<blocks, waves_per_block * 32, 0, stream>>>(
        merge, W_tr, b_tr, lin, acc, deg, out);
  }
}